// Butterfly_61186104098904
// MI455X (gfx1250) — compile-verified
//
#include <hip/hip_runtime.h>
#include <hip/hip_bf16.h>

typedef float v2f __attribute__((ext_vector_type(2)));
typedef float v8f __attribute__((ext_vector_type(8)));

#define N_FULL   1024
#define NBLK     32          // 32 blocks of 32 elements
#define TILE_M   16          // batch rows per workgroup tile
#define YS       1028        // LDS row stride in floats (pad: 4*lane bank step, 16B pitch)
#define NWAVES   4

// ---------------------------------------------------------------------------
// Setup kernel: compose the 10 sparse butterfly stages into 32 dense A_g
// (stages 0-4, intra-block) and 32 dense B_r (stages 5-9, inter-block)
// 32x32 fp32 matrices, and store them pre-packed in WMMA B-operand fragment
// order:  pk[mat*1024 + ((h*8+kk)*32 + lane)*2 + v] = M[h*16 + lane%16][kk*4 + (lane/16)*2 + v]
// so the main kernel's fragment loads are fully coalesced.
// blockIdx.x 0..31 -> A_g (g = bid); 32..63 -> B_r (r = bid-32).
// ---------------------------------------------------------------------------
__global__ __launch_bounds__(32) void build_packed_mats(
    const float* __restrict__ tw,   // (1, 10, 512, 2, 2)
    float* __restrict__ pk)         // 64 * 1024 floats
{
    __shared__ float Cm[32 * 32];
    const int c   = threadIdx.x;            // this lane owns column c
    const bool isB = (blockIdx.x >= 32);
    const int idx = isB ? (int)blockIdx.x - 32 : (int)blockIdx.x;  // g or r

    for (int i = 0; i < 32; ++i) Cm[i * 32 + c] = (i == c) ? 1.0f : 0.0f;
    __syncthreads();

    for (int lsi = 0; lsi < 5; ++lsi) {
        const int ls = isB ? lsi + 5 : lsi;   // log2(stride)
        const int s  = 1 << ls;               // element stride
        const int cs = isB ? (s >> 5) : s;    // stride in this group's coordinate
        // 16 pairs per stage; columns are independent -> no barrier needed
        for (int pr = 0; pr < 16; ++pr) {
            const int G  = pr / cs;
            const int t  = pr % cs;
            const int u0 = G * 2 * cs + t;    // coordinate of lower partner
            const int u1 = u0 + cs;
            const int p0 = isB ? (u0 * 32 + idx) : (idx * 32 + u0);  // element idx
            const int pid = (p0 / (2 * s)) * s + (p0 % s);           // twiddle pair id
            const float* tp = tw + ls * 2048 + pid * 4;
            const float t00 = tp[0], t01 = tp[1], t10 = tp[2], t11 = tp[3];
            const float a0 = Cm[u0 * 32 + c];
            const float a1 = Cm[u1 * 32 + c];
            Cm[u0 * 32 + c] = t00 * a0 + t01 * a1;
            Cm[u1 * 32 + c] = t10 * a0 + t11 * a1;
        }
    }
    __syncthreads();

    // pack WMMA B-operand fragments (coalesced writes)
    const int L = c, Lm = L & 15, Lh = L >> 4;
    float* base = pk + (size_t)blockIdx.x * 1024;
    for (int h = 0; h < 2; ++h) {
        for (int kk = 0; kk < 8; ++kk) {
            const int row  = h * 16 + Lm;
            const int col0 = kk * 4 + Lh * 2;
            base[(h * 8 + kk) * 64 + L * 2 + 0] = Cm[row * 32 + col0 + 0];
            base[(h * 8 + kk) * 64 + L * 2 + 1] = Cm[row * 32 + col0 + 1];
        }
    }
}

// ---------------------------------------------------------------------------
// Main kernel: one 16-row tile per workgroup; all 10 stages fused on-chip.
// Global traffic: 64KB in + 64KB out per WG (bandwidth-optimal).
// ---------------------------------------------------------------------------
__global__ __launch_bounds__(NWAVES * 32) void butterfly_wmma(
    const float* __restrict__ x,
    const float* __restrict__ pk,
    const float* __restrict__ bias,
    float* __restrict__ out)
{
    __shared__ __align__(16) float y[TILE_M * YS];   // 65792 B

    const int tid = threadIdx.x;
    const int w   = tid >> 5;          // wave id 0..3
    const int L   = tid & 31;          // lane
    const int Lm  = L & 15;
    const int Lh  = L >> 4;
    const size_t b0 = (size_t)blockIdx.x * TILE_M;

    // ---- stage in: coalesced global -> LDS -----------------------------
    const float* xrow = x + b0 * N_FULL;
    for (int i = tid; i < TILE_M * (N_FULL / 4); i += NWAVES * 32) {
        const int row = i >> 8;
        const int c4  = (i & 255) * 4;
        const float4 v = *(const float4*)(xrow + row * N_FULL + c4);
        *(float4*)(&y[row * YS + c4]) = v;
    }
    __syncthreads();

    const v2f* pkA = (const v2f*)pk;               // 32 mats * 512 v2f
    const v2f* pkB = (const v2f*)pk + 32 * 512;

    // ---- group A: per block g, Y[b, g*32+i] = sum_j A_g[i,j] X[b, g*32+j]
    for (int g = w; g < NBLK; g += NWAVES) {
        v2f xa[8];
        const int xb = Lm * YS + g * 32 + Lh * 2;
        #pragma unroll
        for (int kk = 0; kk < 8; ++kk)
            xa[kk] = *(const v2f*)(&y[xb + kk * 4]);   // conflict-free ds_load_b64

        #pragma unroll
        for (int h = 0; h < 2; ++h) {
            v8f acc = {0.f, 0.f, 0.f, 0.f, 0.f, 0.f, 0.f, 0.f};
            #pragma unroll
            for (int kk = 0; kk < 8; ++kk) {
                const v2f bm = pkA[g * 512 + (h * 8 + kk) * 32 + L];  // coalesced L2
                acc = __builtin_amdgcn_wmma_f32_16x16x4_f32(
                    false, xa[kk], false, bm, (short)0, acc, false, false);
            }
            const int db = g * 32 + h * 16 + Lm;
            #pragma unroll
            for (int v = 0; v < 8; ++v)
                y[(v + 8 * Lh) * YS + db] = acc[v];    // in-place, block-local
        }
    }
    __syncthreads();

    // ---- group B: per offset r, Z[b, g'*32+r] = sum_g B_r[g',g] Y[b, g*32+r]
    for (int r = w; r < 32; r += NWAVES) {
        v2f ya[8];
        #pragma unroll
        for (int kk = 0; kk < 8; ++kk) {
            const int k0 = kk * 4 + Lh * 2;
            ya[kk][0] = y[Lm * YS + (k0 + 0) * 32 + r];
            ya[kk][1] = y[Lm * YS + (k0 + 1) * 32 + r];
        }
        #pragma unroll
        for (int h = 0; h < 2; ++h) {
            v8f acc = {0.f, 0.f, 0.f, 0.f, 0.f, 0.f, 0.f, 0.f};
            #pragma unroll
            for (int kk = 0; kk < 8; ++kk) {
                const v2f bm = pkB[r * 512 + (h * 8 + kk) * 32 + L];
                acc = __builtin_amdgcn_wmma_f32_16x16x4_f32(
                    false, ya[kk], false, bm, (short)0, acc, false, false);
            }
            #pragma unroll
            for (int v = 0; v < 8; ++v)
                y[(v + 8 * Lh) * YS + (h * 16 + Lm) * 32 + r] = acc[v];  // in-place per r
        }
    }
    __syncthreads();

    // ---- write out: LDS -> global, coalesced, + bias -------------------
    float* orow = out + b0 * N_FULL;
    for (int i = tid; i < TILE_M * (N_FULL / 4); i += NWAVES * 32) {
        const int row = i >> 8;
        const int c4  = (i & 255) * 4;
        float4 v = *(const float4*)(&y[row * YS + c4]);
        const float4 bv = *(const float4*)(bias + c4);
        v.x += bv.x; v.y += bv.y; v.z += bv.z; v.w += bv.w;
        *(float4*)(orow + row * N_FULL + c4) = v;
    }
}

// ---------------------------------------------------------------------------
extern "C" void kernel_launch(void* const* d_in, const int* in_sizes, int n_in,
                              void* d_out, int out_size, void* d_ws, size_t ws_size,
                              hipStream_t stream) {
    (void)n_in; (void)out_size; (void)ws_size;
    const float* x    = (const float*)d_in[0];   // (32768, 1024) fp32
    const float* tw   = (const float*)d_in[1];   // (1, 10, 512, 2, 2) fp32
    const float* bias = (const float*)d_in[2];   // (1024,) fp32
    float* out = (float*)d_out;
    float* pk  = (float*)d_ws;                   // 64 * 1024 floats = 256 KB scratch

    const int batch = in_sizes[0] / N_FULL;

    build_packed_mats<<<64, 32, 0, stream>>>(tw, pk);
    butterfly_wmma<<<batch / TILE_M, NWAVES * 32, 0, stream>>>(x, pk, bias, out);
}